// MultipleNegativesRankingLoss_26817775796655
// MI455X (gfx1250) — compile-verified
//
#include <hip/hip_runtime.h>

// Problem constants (from reference setup_inputs)
#define BB   32
#define NN   8
#define LQ   256
#define LD   512
#define DD   768

#define WAVES_PER_BLOCK 6
#define THREADS_PER_BLOCK (WAVES_PER_BLOCK * 32)            // 192
#define COLS_PER_WAVE   64                                  // 4 interleaved WMMA chains
#define COLS_PER_BLOCK  (WAVES_PER_BLOCK * COLS_PER_WAVE)   // 384
#define CHUNKS          (DD / COLS_PER_BLOCK)               // 2

typedef __attribute__((ext_vector_type(2))) float v2f;
typedef __attribute__((ext_vector_type(4))) float v4f;
typedef __attribute__((ext_vector_type(8))) float v8f;

// ---------------------------------------------------------------------------
// Kernel 0: zero the scalar loss output (docs kernel accumulates atomically).
// ---------------------------------------------------------------------------
__global__ void zero_out_kernel(float* out) {
    if (threadIdx.x == 0) out[0] = 0.0f;
}

// ---------------------------------------------------------------------------
// Column-sum of a ROWS x DD fp32 matrix over 4 rows/iter via chained
// V_WMMA_F32_16X16X4_F32 with an all-ones A: D = ones(16x4) x B(4x16) + C.
// Each wave owns 64 columns as 4 interleaved chains; lane l's b128 load
// covers physical columns 4*(l&15)..4*(l&15)+3, component j feeds chain j.
// Lanes 0-15 carry rows {r, r+1}, lanes 16-31 rows {r+2, r+3} (row placement
// is irrelevant because A is all-ones -> only the K-sum matters).
// ---------------------------------------------------------------------------
template <int ROWS>
__device__ inline void colsum4(const float* __restrict__ base, int cbase,
                               int lane, v8f& a0, v8f& a1, v8f& a2, v8f& a3) {
    const int rowsel = lane >> 4;              // 0: rows r,r+1   1: rows r+2,r+3
    const float* lanebase = base + (size_t)(2 * rowsel) * DD + cbase + 4 * (lane & 15);

    v2f ones; ones[0] = 1.0f; ones[1] = 1.0f;

#pragma unroll 2
    for (int r = 0; r < ROWS; r += 4) {
        v4f ld0 = *(const v4f*)(lanebase + (size_t)r * DD);        // row r   (or r+2)
        v4f ld1 = *(const v4f*)(lanebase + (size_t)r * DD + DD);   // row r+1 (or r+3)
        v2f b0; b0[0] = ld0[0]; b0[1] = ld1[0];
        v2f b1; b1[0] = ld0[1]; b1[1] = ld1[1];
        v2f b2; b2[0] = ld0[2]; b2[1] = ld1[2];
        v2f b3; b3[0] = ld0[3]; b3[1] = ld1[3];
        a0 = __builtin_amdgcn_wmma_f32_16x16x4_f32(false, ones, false, b0, (short)0, a0, false, false);
        a1 = __builtin_amdgcn_wmma_f32_16x16x4_f32(false, ones, false, b1, (short)0, a1, false, false);
        a2 = __builtin_amdgcn_wmma_f32_16x16x4_f32(false, ones, false, b2, (short)0, a2, false, false);
        a3 = __builtin_amdgcn_wmma_f32_16x16x4_f32(false, ones, false, b3, (short)0, a3, false, false);
    }
}

// ---------------------------------------------------------------------------
// Kernel 1: questions column sums -> qsum[b, d] (raw sums; scale folded later)
// Grid: BB * CHUNKS blocks of 192 threads.
// ---------------------------------------------------------------------------
__global__ void __launch_bounds__(THREADS_PER_BLOCK)
qsum_kernel(const float* __restrict__ q, float* __restrict__ qsum) {
    const int blk   = blockIdx.x;
    const int chunk = blk % CHUNKS;
    const int b     = blk / CHUNKS;
    const int wave  = threadIdx.x >> 5;
    const int lane  = threadIdx.x & 31;
    const int cbase = chunk * COLS_PER_BLOCK + wave * COLS_PER_WAVE;

    const float* __restrict__ base = q + (size_t)b * LQ * DD;

    v8f a0 = {}, a1 = {}, a2 = {}, a3 = {};
    colsum4<LQ>(base, cbase, lane, a0, a1, a2, a3);

    // Every row of each D tile equals the column sum; lane l (0..15), chain j
    // holds the sum for physical column cbase + 4*l + j -> one b128 store.
    if (lane < 16) {
        v4f outv; outv[0] = a0[0]; outv[1] = a1[0]; outv[2] = a2[0]; outv[3] = a3[0];
        *(v4f*)(qsum + (size_t)b * DD + cbase + 4 * lane) = outv;
    }
}

// ---------------------------------------------------------------------------
// Kernel 2: docs column sums via WMMA chains, dotted against qsum with the
// identical column permutation, signed, block-reduced, one atomic per block.
// Grid: BB * NN * CHUNKS blocks of 192 threads.
// ---------------------------------------------------------------------------
__global__ void __launch_bounds__(THREADS_PER_BLOCK)
docs_kernel(const float* __restrict__ docs, const unsigned char* __restrict__ is_ans,
            const float* __restrict__ qsum, float* __restrict__ out) {
    const int blk   = blockIdx.x;
    const int chunk = blk % CHUNKS;
    const int n     = (blk / CHUNKS) % NN;
    const int b     = blk / (CHUNKS * NN);
    const int wave  = threadIdx.x >> 5;
    const int lane  = threadIdx.x & 31;
    const int cbase = chunk * COLS_PER_BLOCK + wave * COLS_PER_WAVE;

    const float* __restrict__ base = docs + ((size_t)(b * NN + n)) * LD * DD;

    v8f a0 = {}, a1 = {}, a2 = {}, a3 = {};
    colsum4<LD>(base, cbase, lane, a0, a1, a2, a3);

    // Same permuted column mapping for qsum: lane l, chain j <-> col 4*l + j.
    v4f q4 = *(const v4f*)(qsum + (size_t)b * DD + cbase + 4 * (lane & 15));
    float val = 0.0f;
    if (lane < 16)   // lanes 16..31 hold replicated rows -> would double count
        val = a0[0] * q4[0] + a1[0] * q4[1] + a2[0] * q4[2] + a3[0] * q4[3];

    // wave32 butterfly reduction
#pragma unroll
    for (int off = 16; off > 0; off >>= 1)
        val += __shfl_xor(val, off, 32);

    __shared__ float red[WAVES_PER_BLOCK];
    if (lane == 0) red[wave] = val;
    __syncthreads();

    if (threadIdx.x == 0) {
        float s = 0.0f;
#pragma unroll
        for (int w = 0; w < WAVES_PER_BLOCK; ++w) s += red[w];
        const float sign  = is_ans[b * NN + n] ? -1.0f : 1.0f;
        const float scale = sign * (1.0f / ((float)LQ * (float)LD));
        atomicAdd(out, s * scale);   // global_atomic_add_f32
    }
}

// ---------------------------------------------------------------------------
extern "C" void kernel_launch(void* const* d_in, const int* in_sizes, int n_in,
                              void* d_out, int out_size, void* d_ws, size_t ws_size,
                              hipStream_t stream) {
    const float*         questions = (const float*)d_in[0];         // [B, Lq, D]
    const float*         docs      = (const float*)d_in[1];         // [B, N, Ld, D]
    const unsigned char* is_ans    = (const unsigned char*)d_in[2]; // [B, N] bool (1 byte)
    float*               out       = (float*)d_out;                 // scalar loss

    float* qsum = (float*)d_ws;  // BB*DD floats = 96 KiB of workspace

    zero_out_kernel<<<1, 64, 0, stream>>>(out);
    qsum_kernel<<<BB * CHUNKS, THREADS_PER_BLOCK, 0, stream>>>(questions, qsum);
    docs_kernel<<<BB * NN * CHUNKS, THREADS_PER_BLOCK, 0, stream>>>(docs, is_ans, qsum, out);
    (void)in_sizes; (void)n_in; (void)out_size; (void)ws_size;
}